// Conv2dExponentialDelayed_38491496907210
// MI455X (gfx1250) — compile-verified
//
#include <hip/hip_runtime.h>

// Geometry (fixed by the reference)
#define B_     4
#define CIN_   8
#define HW_    64
#define COUT_  32
#define K_     72
#define L_     4096      // HO*WO = 64*64
#define LT_    16        // l-positions per wave tile
#define NBO_   (B_ * COUT_)      // 128
#define WAVES_PER_BLOCK 8
#define NLTG_  (L_ / (LT_ * WAVES_PER_BLOCK))   // 32 l-groups
#define NBLOCKS (NBO_ * NLTG_)                  // 4096

typedef float v2f __attribute__((ext_vector_type(2)));
typedef float v8f __attribute__((ext_vector_type(8)));

// One block = one (b,o) [block-uniform -> SGPR bases, saddr+voffset addressing]
// and 8 adjacent 16-wide l-tiles (one per wave). Per wave, k runs in chunks of 4
// mapped on V_WMMA_F32_16X16X4_F32 (A row M=0 = weights, B = spikes,
// D[0][n] = pot). Wave32 32-bit matrix layouts (ISA 7.12.2):
//   A: lanes 0-15 -> M=lane, VGPR{0,1}=K{0,1}; lanes 16-31 -> K{2,3}
//   B: lanes 0-15 -> N=lane, VGPR{0,1}=K{0,1}; lanes 16-31 -> N=lane-16, K{2,3}
//   D: VGPR0, lanes 0-15 = (M=0, N=lane)
// Lane handles k = kb + 2*half + v. Affine-in-half parts are folded into the
// per-lane voffset once; k->(c,kh,kw) is decomposed at COMPILE TIME for both
// half values (full unroll) and resolved with one cndmask on precomputed
// clamped offsets + one cndmask on precomputed {0,1} float masks (no i1 selects).
__global__ __launch_bounds__(256)
void Conv2dExponentialDelayed_kernel(const float* __restrict__ x,
                                     const float* __restrict__ weight,
                                     const float* __restrict__ trace,
                                     const float* __restrict__ delay,
                                     const float* __restrict__ delay_init,
                                     const float* __restrict__ p_alpha,
                                     const float* __restrict__ p_tau,
                                     const float* __restrict__ p_dt,
                                     float* __restrict__ out_pot,
                                     float* __restrict__ out_trace)
{
    const int lane = threadIdx.x & 31;
    const int wave = threadIdx.x >> 5;

    // Block-uniform work assignment
    const int bo  = blockIdx.x & (NBO_ - 1);   // uniform: b*32 + o
    const int ltg = blockIdx.x >> 7;           // uniform: l-group 0..31
    const int b   = bo >> 5;
    const int o   = bo & (COUT_ - 1);

    const int n    = lane & 15;     // WMMA column == l offset in tile
    const int half = lane >> 4;     // k-pair selector within a chunk
    const int lt   = ltg * WAVES_PER_BLOCK + wave;
    const int l    = lt * LT_ + n;
    const int ho   = l >> 6;        // wave-uniform (LT_ divides 64)
    const int wo   = l & 63;

    const float alpha = p_alpha[0];
    const float rr    = p_dt[0] / p_tau[0];   // dt / tau

    // Uniform (SGPR) bases; per-lane work lives in one 32-bit voffset each.
    const size_t base_bo = (size_t)bo * (size_t)(K_ * L_);
    const float* __restrict__ trb = trace      + base_bo;
    const float* __restrict__ dlb = delay      + base_bo;
    const float* __restrict__ dib = delay_init + base_bo;
    float*       __restrict__ otb = out_trace  + base_bo;
    const float* __restrict__ wrb = weight + (size_t)o * K_;
    const float* __restrict__ xbb = x + (size_t)b * (CIN_ * HW_ * HW_);

    const int vo  = 2 * half * L_ + l;  // stream voffset (elements); k part is a literal
    const int wvo = 2 * half;           // weight voffset

    // Clamped tap offsets (always in-bounds) and {0,1} float padding masks.
    int   pabs[3][3];
    float mf[3][3];
    {
        int   hrow[3]; float hokf[3];
        int   wcl[3];  float wokf[3];
        #pragma unroll
        for (int kh = 0; kh < 3; ++kh) {
            const int h = ho + kh - 1;                       // wave-uniform
            hrow[kh] = (h < 0 ? 0 : (h > HW_ - 1 ? HW_ - 1 : h)) * HW_;
            hokf[kh] = ((unsigned)h < (unsigned)HW_) ? 1.0f : 0.0f;
        }
        #pragma unroll
        for (int kw = 0; kw < 3; ++kw) {
            const int w = wo + kw - 1;
            wcl[kw]  = (w < 0 ? 0 : (w > HW_ - 1 ? HW_ - 1 : w));
            wokf[kw] = ((unsigned)w < (unsigned)HW_) ? 1.0f : 0.0f;
        }
        #pragma unroll
        for (int kh = 0; kh < 3; ++kh)
            #pragma unroll
            for (int kw = 0; kw < 3; ++kw) {
                pabs[kh][kw] = hrow[kh] + wcl[kw];
                mf[kh][kw]   = hokf[kh] * wokf[kw];
            }
    }
    const float mrow0 = (n == 0) ? 1.0f : 0.0f;   // A-matrix row-0 mask

    v8f acc = {0.f, 0.f, 0.f, 0.f, 0.f, 0.f, 0.f, 0.f};

    #pragma unroll
    for (int kb = 0; kb < K_; kb += 4) {
        v2f am, bm;
        #pragma unroll
        for (int v = 0; v < 2; ++v) {
            // Candidate k's: half=0 -> ka, half=1 -> kc (constants after unroll)
            const int ka = kb + v;
            const int kc = kb + v + 2;
            const int ca = ka / 9, ra = ka % 9, kha = ra / 3, kwa = ra % 3;
            const int cc = kc / 9, rc = kc % 9, khc = rc / 3, kwc = rc % 3;

            // One cndmask for the (clamped, in-bounds) tap offset, one for the
            // padding mask; channel constant folds into the load literal.
            const int off = half ? (pabs[khc][kwc] + (cc - ca) * (HW_ * HW_))
                                 : pabs[kha][kwa];
            const float mfv = half ? mf[khc][kwc] : mf[kha][kwa];
            const float xv  = mfv * xbb[ca * (HW_ * HW_) + off];  // L2-resident x

            // Single-use 600MB stream: non-temporal, saddr + voffset + literal.
            const int koff = (kb + v) * L_;
            const float tr = __builtin_nontemporal_load(trb + vo + koff);
            const float dl = __builtin_nontemporal_load(dlb + vo + koff);
            const float di = __builtin_nontemporal_load(dib + vo + koff);

            // exponential trace update
            const float tn = tr + rr * (alpha * xv - tr);
            __builtin_nontemporal_store(tn, otb + vo + koff);

            // delay line: spike when loaded counter hits exactly 1 (exact ints in f32)
            const float dset = dl + xv * di;
            bm[v] = (dset == 1.0f) ? 1.0f : 0.0f;
            // A-matrix: only row M==0 carries the weights (cached b64-pair load)
            am[v] = mrow0 * wrb[wvo + kb + v];
        }
        // acc(16x16) += A(16x4) * B(4x16); only row M=0 is meaningful
        acc = __builtin_amdgcn_wmma_f32_16x16x4_f32(
            /*neg_a=*/false, am, /*neg_b=*/false, bm,
            /*c_mod=*/(short)0, acc, /*reuse_a=*/false, /*reuse_b=*/false);
    }

    // D[M=0][N=n] lives in VGPR0 (acc[0]) of lanes 0..15
    if (lane < 16) {
        out_pot[(size_t)bo * L_ + (size_t)l] = acc[0];
    }
}

extern "C" void kernel_launch(void* const* d_in, const int* in_sizes, int n_in,
                              void* d_out, int out_size, void* d_ws, size_t ws_size,
                              hipStream_t stream) {
    (void)in_sizes; (void)n_in; (void)d_ws; (void)ws_size; (void)out_size;

    const float* x          = (const float*)d_in[0];
    const float* weight     = (const float*)d_in[1];
    const float* trace      = (const float*)d_in[2];
    const float* delay      = (const float*)d_in[3];
    const float* delay_init = (const float*)d_in[4];
    const float* alpha_t    = (const float*)d_in[5];
    const float* tau_t      = (const float*)d_in[6];
    const float* dt         = (const float*)d_in[7];

    float* out_pot   = (float*)d_out;                               // [B, COUT, HO, WO]
    float* out_trace = (float*)d_out + (size_t)B_ * COUT_ * L_;     // trace_new flat == traces_folded flat

    Conv2dExponentialDelayed_kernel<<<NBLOCKS, 256, 0, stream>>>(
        x, weight, trace, delay, delay_init, alpha_t, tau_t, dt,
        out_pot, out_trace);
}